// BaseCollectionLoss_22368189677900
// MI455X (gfx1250) — compile-verified
//
#include <hip/hip_runtime.h>
#include <hip/hip_bf16.h>

// CDNA5 / gfx1250: wave32, WMMA f32 16x16x4 used as a wave-level reduction.

typedef __attribute__((ext_vector_type(2))) float v2f;
typedef __attribute__((ext_vector_type(8))) float v8f;

#define REG_FOLD 100.0f
#define EPS_BUF  0.01f

__device__ __forceinline__ float dist2x2(float a00, float a01, float a10, float a11) {
    // || A^T A - I ||_F^2  for 2x2 A
    float f00 = __builtin_fmaf(a00, a00, a10 * a10);
    float f01 = __builtin_fmaf(a00, a01, a10 * a11);
    float f11 = __builtin_fmaf(a01, a01, a11 * a11);
    float d0 = f00 - 1.0f;
    float d1 = f11 - 1.0f;
    return __builtin_fmaf(d0, d0, __builtin_fmaf(2.0f * f01, f01, d1 * d1));
}

__global__ void bcl_init_acc(double* acc) {
    if (threadIdx.x == 0) { acc[0] = 0.0; acc[1] = 0.0; }
}

__global__ __launch_bounds__(256) void bcl_main(
    const float4* __restrict__ Jf, const float4* __restrict__ Jq,
    const float4* __restrict__ Jg, const float2* __restrict__ mf,
    const float2* __restrict__ mq, double* __restrict__ acc, int n)
{
    float pp_acc  = 0.0f;   // sum of masked loss
    float cnt_acc = 0.0f;   // count of in-mask elements

    const int stride = (int)(gridDim.x * blockDim.x);
    for (int i = (int)(blockIdx.x * blockDim.x + threadIdx.x); i < n; i += stride) {
        // float4 layout: x=J[0][0], y=J[0][1], z=J[1][0], w=J[1][1]
        const float4 f = Jf[i];
        const float4 q = Jq[i];
        const float4 g = Jg[i];
        const float2 sf = mf[i];
        const float2 sq = mq[i];

        const bool in_mask =
            (__builtin_fmaf(sf.x, sf.x, sf.y * sf.y) <= 1.0f) &&
            (__builtin_fmaf(sq.x, sq.x, sq.y * sq.y) <= 1.0f);

        const float detg = __builtin_fmaf(g.x, g.w, -(g.y * g.z));
        const float detf = __builtin_fmaf(f.x, f.w, -(f.y * f.z));
        const float detq = __builtin_fmaf(q.x, q.w, -(q.y * q.z));

        const float rg = 1.0f / detg;
        const float rf = 1.0f / detf;

        // inv([x y; z w]) = [w -y; -z x] / det
        const float gi00 =  g.w * rg, gi01 = -g.y * rg;
        const float gi10 = -g.z * rg, gi11 =  g.x * rg;
        const float fi00 =  f.w * rf, fi01 = -f.y * rf;
        const float fi10 = -f.z * rf, fi11 =  f.x * rf;

        // J_fhg = Jf * Jg^-1
        const float A00 = __builtin_fmaf(f.x, gi00, f.y * gi10);
        const float A01 = __builtin_fmaf(f.x, gi01, f.y * gi11);
        const float A10 = __builtin_fmaf(f.z, gi00, f.w * gi10);
        const float A11 = __builtin_fmaf(f.z, gi01, f.w * gi11);
        // J_qhg = Jq * Jg^-1
        const float B00 = __builtin_fmaf(q.x, gi00, q.y * gi10);
        const float B01 = __builtin_fmaf(q.x, gi01, q.y * gi11);
        const float B10 = __builtin_fmaf(q.z, gi00, q.w * gi10);
        const float B11 = __builtin_fmaf(q.z, gi01, q.w * gi11);
        // J_qhf = Jq * Jf^-1
        const float C00 = __builtin_fmaf(q.x, fi00, q.y * fi10);
        const float C01 = __builtin_fmaf(q.x, fi01, q.y * fi11);
        const float C10 = __builtin_fmaf(q.z, fi00, q.w * fi10);
        const float C11 = __builtin_fmaf(q.z, fi01, q.w * fi11);

        const float ppd = dist2x2(A00, A01, A10, A11)
                        + dist2x2(B00, B01, B10, B11)
                        + dist2x2(C00, C01, C10, C11);

        const float fold = REG_FOLD * (fmaxf(0.0f, EPS_BUF - detf) +
                                       fmaxf(0.0f, EPS_BUF - detq));

        if (in_mask) {
            pp_acc  += ppd + fold;
            cnt_acc += 1.0f;
        }
    }

    // ---- wave-level reduction via V_WMMA_F32_16X16X4_F32 ----
    // A (16x4 f32): lane L<16 holds A[L][0]=val, A[L][1]=0; lane L+16 holds
    // A[L][2]=val, A[L][3]=0.  B = all-ones 4x16 -> every D[m][n] = row-sum(m).
    // EXEC is all-ones here (outside divergent flow), as WMMA requires.
    v2f a_pp;  a_pp[0] = pp_acc;  a_pp[1] = 0.0f;
    v2f a_ct;  a_ct[0] = cnt_acc; a_ct[1] = 0.0f;
    v2f ones;  ones[0] = 1.0f;    ones[1] = 1.0f;
    v8f czero = {};

    v8f d_pp = __builtin_amdgcn_wmma_f32_16x16x4_f32(
        false, a_pp, false, ones, (short)0, czero, false, false);
    v8f d_ct = __builtin_amdgcn_wmma_f32_16x16x4_f32(
        false, a_ct, false, ones, (short)0, czero, false, false);

    // C/D layout: VGPR j of lane L<16 = D[j][L]; lane L+16 = D[j+8][L].
    // Column 0 (lanes 0 and 16) covers all 16 rows exactly once.
    float s_pp = 0.0f, s_ct = 0.0f;
    #pragma unroll
    for (int j = 0; j < 8; ++j) { s_pp += d_pp[j]; s_ct += d_ct[j]; }

    const unsigned lane = threadIdx.x & 31u;  // wave32
    if (lane == 0u || lane == 16u) {
        atomicAdd(&acc[0], (double)s_pp);   // global_atomic_add_f64
        atomicAdd(&acc[1], (double)s_ct);
    }
}

__global__ void bcl_finalize(const double* __restrict__ acc, float* __restrict__ out) {
    if (threadIdx.x == 0) {
        out[0] = (float)(acc[0] / acc[1]);
    }
}

extern "C" void kernel_launch(void* const* d_in, const int* in_sizes, int n_in,
                              void* d_out, int out_size, void* d_ws, size_t ws_size,
                              hipStream_t stream) {
    (void)n_in; (void)out_size; (void)ws_size;
    const float4* Jf = (const float4*)d_in[0];
    const float4* Jq = (const float4*)d_in[1];
    const float4* Jg = (const float4*)d_in[2];
    const float2* mf = (const float2*)d_in[3];
    const float2* mq = (const float2*)d_in[4];
    float*  out = (float*)d_out;
    double* acc = (double*)d_ws;

    const int n = in_sizes[0] / 4;          // J_f is N*2*2 floats

    const int threads = 256;                 // 8 wave32s per block
    int blocks = (n + threads * 4 - 1) / (threads * 4);  // ~4 elems/lane
    if (blocks < 1)    blocks = 1;
    if (blocks > 8192) blocks = 8192;

    bcl_init_acc<<<1, 64, 0, stream>>>(acc);
    bcl_main<<<blocks, threads, 0, stream>>>(Jf, Jq, Jg, mf, mq, acc, n);
    bcl_finalize<<<1, 64, 0, stream>>>(acc, out);
}